// IRRTR_a13_5059471475230
// MI455X (gfx1250) — compile-verified
//
#include <hip/hip_runtime.h>
#include <hip/hip_bf16.h>
#include <math.h>

// ---- problem constants (match reference) ----
constexpr int cB = 4, cN = 4096, cK = 16, cK1 = 17, cKP = 8, cC = 3;
constexpr int cS = 64, cS2 = cS * cS, cOUT = 3;
constexpr float cEPS = 1e-8f;
constexpr int cTS = 512;               // LDS point-tile size for KNN
constexpr float cINVS = 1.0f / (cS - 1);

typedef __attribute__((ext_vector_type(2))) float v2f;
typedef __attribute__((ext_vector_type(8))) float v8f;

// ============================================================
// 1) Self-KNN: brute force 4096x4096, keep 17 nearest, drop self.
//    Register-resident unrolled insertion sort, strict '<' so ties
//    keep the lower index (matches jax top_k tie-break).
// ============================================================
__global__ void IRRTR_knn_self(const float* __restrict__ xy, int* __restrict__ idx1) {
    const int b = blockIdx.y;
    const int q = blockIdx.x * blockDim.x + threadIdx.x;       // 0..N-1
    const float* P = xy + (size_t)b * cN * 2;
    const float qx = P[q * 2 + 0], qy = P[q * 2 + 1];

    float bd[cK1]; int bi[cK1];
#pragma unroll
    for (int t = 0; t < cK1; ++t) { bd[t] = 3.4e38f; bi[t] = 0; }

    __shared__ float2 tile[cTS];
    for (int t0 = 0; t0 < cN; t0 += cTS) {
        __syncthreads();
        for (int i = threadIdx.x; i < cTS; i += blockDim.x)
            tile[i] = ((const float2*)P)[t0 + i];
        __syncthreads();
        for (int j = 0; j < cTS; ++j) {
            const float dx = tile[j].x - qx, dy = tile[j].y - qy;
            float d = dx * dx + dy * dy;
            int ji = t0 + j;
            if (d < bd[cK1 - 1]) {
#pragma unroll
                for (int t = 0; t < cK1; ++t) {          // insert keeping ascending order
                    const bool ins = d < bd[t];
                    const float td = ins ? bd[t] : d;
                    const int   ti = ins ? bi[t] : ji;
                    bd[t] = ins ? d : bd[t];
                    bi[t] = ins ? ji : bi[t];
                    d = td; ji = ti;
                }
            }
        }
    }
    int* o = idx1 + ((size_t)b * cN + q) * cK;
#pragma unroll
    for (int t = 1; t < cK1; ++t) o[t - 1] = bi[t];            // drop self (slot 0, d==0)
}

// ============================================================
// 2) IDW gradient: grad[b,n,c*2+d] = (sum_k w^2 * dv_c * rel_d) / sum_k w,
//    w = 1/(|rel|^2 + EPS).  Written directly to the second output.
// ============================================================
__global__ void IRRTR_gradient(const float* __restrict__ xy, const float* __restrict__ v,
                               const int* __restrict__ idx1, float* __restrict__ grad) {
    const int t = blockIdx.x * blockDim.x + threadIdx.x;
    if (t >= cB * cN) return;
    const int b = t / cN, n = t % cN;
    const float* P = xy + (size_t)b * cN * 2;
    const float* V = v + (size_t)b * cN * cC;
    const float qx = P[n * 2], qy = P[n * 2 + 1];
    const float v0 = V[n * 3], v1 = V[n * 3 + 1], v2 = V[n * 3 + 2];
    float a00 = 0, a01 = 0, a10 = 0, a11 = 0, a20 = 0, a21 = 0, wsum = 0;
    const int* nb = idx1 + (size_t)t * cK;
    for (int k = 0; k < cK; ++k) {
        const int j = nb[k];
        const float rx = P[j * 2] - qx, ry = P[j * 2 + 1] - qy;
        const float d2 = rx * rx + ry * ry + cEPS;
        const float w = 1.0f / d2, w2 = w * w;                 // w/d2 == w^2
        const float dv0 = V[j * 3] - v0, dv1 = V[j * 3 + 1] - v1, dv2 = V[j * 3 + 2] - v2;
        a00 += w2 * dv0 * rx; a01 += w2 * dv0 * ry;
        a10 += w2 * dv1 * rx; a11 += w2 * dv1 * ry;
        a20 += w2 * dv2 * rx; a21 += w2 * dv2 * ry;
        wsum += w;
    }
    const float inv = 1.0f / wsum;
    float* g = grad + (size_t)t * 6;
    g[0] = a00 * inv; g[1] = a01 * inv; g[2] = a10 * inv;
    g[3] = a11 * inv; g[4] = a20 * inv; g[5] = a21 * inv;
}

// ============================================================
// 3) Angular sort + build feat_in [B*N*K, 16]:
//    cols 0-4 = sorted [rel, dv], 5-9 = roll(-1) of same, 10-15 = sorted neighbor gradient
// ============================================================
__global__ void IRRTR_build_feat(const float* __restrict__ xy, const float* __restrict__ v,
                                 const int* __restrict__ idx1, const float* __restrict__ grad,
                                 float* __restrict__ feat) {
    const int t = blockIdx.x * blockDim.x + threadIdx.x;
    if (t >= cB * cN) return;
    const int b = t / cN, n = t % cN;
    const float* P = xy + (size_t)b * cN * 2;
    const float* V = v + (size_t)b * cN * cC;
    const float qx = P[n * 2], qy = P[n * 2 + 1];
    const float v0 = V[n * 3], v1 = V[n * 3 + 1], v2 = V[n * 3 + 2];
    const int* nb = idx1 + (size_t)t * cK;

    float ang[cK];
    for (int k = 0; k < cK; ++k) {
        const int j = nb[k];
        ang[k] = atan2f(P[j * 2 + 1] - qy, P[j * 2] - qx);
    }
    // stable selection-sort ascending (lower original index wins ties)
    int ord[cK]; bool used[cK];
    for (int k = 0; k < cK; ++k) used[k] = false;
    for (int s = 0; s < cK; ++s) {
        int best = 0; float ba = 3.4e38f;
        for (int k = 0; k < cK; ++k)
            if (!used[k] && ang[k] < ba) { ba = ang[k]; best = k; }
        used[best] = true; ord[s] = best;
    }
    float* frow = feat + (size_t)t * cK * 16;
    for (int s = 0; s < cK; ++s) {
        const int j = nb[ord[s]];
        const float rx = P[j * 2] - qx, ry = P[j * 2 + 1] - qy;
        const float d0 = V[j * 3] - v0, d1 = V[j * 3 + 1] - v1, d2v = V[j * 3 + 2] - v2;
        float* r0 = frow + s * 16;
        r0[0] = rx; r0[1] = ry; r0[2] = d0; r0[3] = d1; r0[4] = d2v;
        float* rp = frow + ((s - 1 + cK) & (cK - 1)) * 16;     // this entry is row (s-1)'s "next"
        rp[5] = rx; rp[6] = ry; rp[7] = d0; rp[8] = d1; rp[9] = d2v;
        const float* g = grad + ((size_t)b * cN + j) * 6;
        r0[10] = g[0]; r0[11] = g[1]; r0[12] = g[2]; r0[13] = g[3]; r0[14] = g[4]; r0[15] = g[5];
    }
}

// ============================================================
// 4) fp32 WMMA GEMM, weight-stationary: D = pool(relu(A[M,Kd] @ W[Kd,N] + bias))
//    Each wave owns one 16-col strip (nt): it hoists ALL B fragments of
//    that strip into registers ONCE (Kd/4 x v2f, <= 72 VGPRs for Kd=144),
//    then grid-strides over 16-row M-tiles. Inner loop = 1 global_load_b64
//    (A pair) + 1 v_wmma_f32_16x16x4_f32 per k-step, fully unrolled.
//    pool in {1,8,16}: max-reduce over row groups inside the D fragments
//    (pool==16 fuses the stage-2 neighbor max, pool==8 the stage-3 one).
// ============================================================
template <int Kd>
__global__ void IRRTR_wmma_gemm(const float* __restrict__ A, const float* __restrict__ W,
                                const float* __restrict__ bias, float* __restrict__ D,
                                int M, int Ncol, int pool, int doRelu) {
    constexpr int KS = Kd / 4;                   // k-steps
    const int ntiles = Ncol >> 4;
    const int mtiles = M >> 4;
    const int nwaves = gridDim.x * (blockDim.x >> 5);          // multiple of ntiles by launch
    const int wave = blockIdx.x * (blockDim.x >> 5) + (threadIdx.x >> 5);
    const int lane = threadIdx.x & 31;
    const int half = lane >> 4;                  // 0: K offsets {0,1}; 1: {2,3}
    const int l16 = lane & 15;

    const int nt = wave % ntiles;                // this wave's column strip (stationary)
    const int n0 = nt << 4;
    const int mstride = nwaves / ntiles;

    // ---- hoist the whole B strip into registers (read once per wave) ----
    v2f bfr[KS];
#pragma unroll
    for (int s = 0; s < KS; ++s) {               // B: row K=4s+2*half+{0,1}, col n0+l16
        const float* bp = W + (size_t)(4 * s + half * 2) * Ncol + n0 + l16;
        bfr[s].x = bp[0];
        bfr[s].y = bp[Ncol];
    }
    const float bb = bias ? bias[n0 + l16] : 0.0f;

    for (int mt = wave / ntiles; mt < mtiles; mt += mstride) { // wave-uniform loop
        const int m0 = mt << 4;
        const float* arow = A + (size_t)(m0 + l16) * Kd + half * 2;
        // prefetch next M-tile's A row into near cache
        __builtin_prefetch(arow + (size_t)mstride * 16 * Kd, 0, 3);

        v8f acc = {};
#pragma unroll
        for (int s = 0; s < KS; ++s) {           // A: row m0+l16, K=4s+2*half+{0,1}
            v2f a;
            a.x = arow[4 * s];
            a.y = arow[4 * s + 1];
            acc = __builtin_amdgcn_wmma_f32_16x16x4_f32(
                      false, a, false, bfr[s], (short)0, acc, false, false);
        }

        float vals[8];
#pragma unroll
        for (int r = 0; r < 8; ++r) {
            float x = acc[r] + bb;
            vals[r] = doRelu ? fmaxf(x, 0.0f) : x;
        }
        if (pool == 1) {
#pragma unroll
            for (int r = 0; r < 8; ++r)          // D VGPR r -> row m0 + r + 8*half, col n0+l16
                D[(size_t)(m0 + r + half * 8) * Ncol + n0 + l16] = vals[r];
        } else if (pool == 8) {                  // rows 0-7 live in half 0, 8-15 in half 1
            float mx = vals[0];
#pragma unroll
            for (int r = 1; r < 8; ++r) mx = fmaxf(mx, vals[r]);
            D[(size_t)((m0 >> 3) + half) * Ncol + n0 + l16] = mx;
        } else {                                 // pool == 16: also combine the two lane halves
            float mx = vals[0];
#pragma unroll
            for (int r = 1; r < 8; ++r) mx = fmaxf(mx, vals[r]);
            mx = fmaxf(mx, __shfl_xor(mx, 16, 32));
            if (half == 0)
                D[(size_t)(m0 >> 4) * Ncol + n0 + l16] = mx;
        }
    }
}

// helper: pick a wave count that is a multiple of ntiles and covers the tiles
static inline int gemm_blocks(int M, int Ncol) {
    const int ntiles = Ncol >> 4;
    const int tiles = (M >> 4) * ntiles;
    int nwaves = tiles < 4096 ? tiles : 4096;    // 4096 is a multiple of 4 and 8
    // ensure multiple of ntiles (tiles always is; 4096 is for ntiles in {4,8})
    nwaves -= nwaves % ntiles;
    if (nwaves < ntiles) nwaves = ntiles;
    return (nwaves + 7) / 8;                     // 8 waves per 256-thread block
}

// ============================================================
// 5) Grid KNN (KP=8 nearest sample points per 64x64 grid node)
// ============================================================
__global__ void IRRTR_knn_grid(const float* __restrict__ xy, int* __restrict__ idx2) {
    const int b = blockIdx.y;
    const int m = blockIdx.x * blockDim.x + threadIdx.x;       // 0..S2-1
    const float qx = (float)(m / cS) * cINVS;
    const float qy = (float)(m % cS) * cINVS;
    const float* P = xy + (size_t)b * cN * 2;

    float bd[cKP]; int bi[cKP];
#pragma unroll
    for (int t = 0; t < cKP; ++t) { bd[t] = 3.4e38f; bi[t] = 0; }

    __shared__ float2 tile[cTS];
    for (int t0 = 0; t0 < cN; t0 += cTS) {
        __syncthreads();
        for (int i = threadIdx.x; i < cTS; i += blockDim.x)
            tile[i] = ((const float2*)P)[t0 + i];
        __syncthreads();
        for (int j = 0; j < cTS; ++j) {
            const float dx = tile[j].x - qx, dy = tile[j].y - qy;
            float d = dx * dx + dy * dy;
            int ji = t0 + j;
            if (d < bd[cKP - 1]) {
#pragma unroll
                for (int t = 0; t < cKP; ++t) {
                    const bool ins = d < bd[t];
                    const float td = ins ? bd[t] : d;
                    const int   ti = ins ? bi[t] : ji;
                    bd[t] = ins ? d : bd[t];
                    bi[t] = ins ? ji : bi[t];
                    d = td; ji = ti;
                }
            }
        }
    }
    int* o = idx2 + ((size_t)b * cS2 + m) * cKP;
#pragma unroll
    for (int t = 0; t < cKP; ++t) o[t] = bi[t];
}

// ============================================================
// 6) Build cat [B*S2*KP, 144] = [qv(3), qg(6), qf(128), rel2(2), dist(1), taylor(3), ang(1)]
// ============================================================
__global__ void IRRTR_build_cat(const float* __restrict__ xy, const float* __restrict__ v,
                                const float* __restrict__ grad, const float* __restrict__ feat,
                                const int* __restrict__ idx2, float* __restrict__ cat) {
    const int t = blockIdx.x * blockDim.x + threadIdx.x;
    if (t >= cB * cS2 * cKP) return;
    const int b = t / (cS2 * cKP);
    const int rem = t - b * (cS2 * cKP);
    const int m = rem / cKP;
    const int j = idx2[t];
    const float gx = (float)(m / cS) * cINVS;
    const float gy = (float)(m % cS) * cINVS;
    const float* P = xy + (size_t)b * cN * 2;
    const float* V = v + (size_t)b * cN * cC;
    const float* G = grad + ((size_t)b * cN + j) * 6;
    const float* F = feat + ((size_t)b * cN + j) * 128;

    const float rx = P[j * 2] - gx, ry = P[j * 2 + 1] - gy;
    const float dist = sqrtf(rx * rx + ry * ry + cEPS);
    const float ang = atan2f(ry, rx);

    float* r = cat + (size_t)t * 144;
    const float q0 = V[j * 3], q1 = V[j * 3 + 1], q2 = V[j * 3 + 2];
    r[0] = q0; r[1] = q1; r[2] = q2;
#pragma unroll
    for (int c = 0; c < 6; ++c) r[3 + c] = G[c];
    for (int c = 0; c < 128; ++c) r[9 + c] = F[c];
    r[137] = rx; r[138] = ry; r[139] = dist;
    r[140] = q0 + G[0] * rx + G[1] * ry;          // taylor = qv + qg[c,:] . rel2
    r[141] = q1 + G[2] * rx + G[3] * ry;
    r[142] = q2 + G[4] * rx + G[5] * ry;
    r[143] = ang;
}

// ============================================================
// 7) Final 128 -> 3 projection + transpose to [B,3,S,S]
// ============================================================
__global__ void IRRTR_final(const float* __restrict__ h, const float* __restrict__ cw2,
                            const float* __restrict__ cb2, float* __restrict__ out) {
    const int t = blockIdx.x * blockDim.x + threadIdx.x;
    if (t >= cB * cS2 * cOUT) return;
    const int m = t / cOUT, o = t - (t / cOUT) * cOUT;
    const float* hr = h + (size_t)m * 128;
    float s = cb2[o];
    for (int j = 0; j < 128; ++j) s += hr[j] * cw2[j * cOUT + o];
    const int b = m / cS2, rm = m % cS2, i = rm / cS, jj = rm % cS;
    out[(((size_t)b * cOUT + o) * cS + i) * cS + jj] = s;
}

// ============================================================
// Launch
// ============================================================
extern "C" void kernel_launch(void* const* d_in, const int* in_sizes, int n_in,
                              void* d_out, int out_size, void* d_ws, size_t ws_size,
                              hipStream_t stream) {
    const float* xy    = (const float*)d_in[0];
    const float* val   = (const float*)d_in[1];
    const float* sf_w1 = (const float*)d_in[2];
    const float* sf_b1 = (const float*)d_in[3];
    const float* sf_w2 = (const float*)d_in[4];
    const float* sf_b2 = (const float*)d_in[5];
    const float* sf_w3 = (const float*)d_in[6];
    const float* sf_b3 = (const float*)d_in[7];
    const float* rt_w1 = (const float*)d_in[8];
    const float* rt_b1 = (const float*)d_in[9];
    const float* cw1   = (const float*)d_in[10];
    const float* cb1   = (const float*)d_in[11];
    const float* cw2   = (const float*)d_in[12];
    const float* cb2   = (const float*)d_in[13];

    float* out  = (float*)d_out;                              // [B,3,S,S] = 49152 floats
    float* grad = out + (size_t)cB * cOUT * cS * cS;          // [B,N,6]   = 98304 floats

    // ---- workspace carve-out (~162 MB, buffers reused) ----
    char* w = (char*)d_ws;
    size_t off = 0;
    auto carve = [&](size_t bytes) -> void* {
        void* p = w + off;
        off += (bytes + 255) & ~(size_t)255;
        return p;
    };
    int*   idx1    = (int*)  carve((size_t)cB * cN * cK * 4);          // 1 MB
    int*   idx2    = (int*)  carve((size_t)cB * cS2 * cKP * 4);        // 0.5 MB
    float* feature = (float*)carve((size_t)cB * cN * 128 * 4);         // 8 MB
    float* h2pool  = (float*)carve((size_t)cB * cS2 * 128 * 4);        // 8 MB
    float* h3b     = (float*)carve((size_t)cB * cS2 * 128 * 4);        // 8 MB
    float* bufA    = (float*)carve((size_t)cB * cS2 * cKP * 144 * 4);  // 72 MB: feat_in / h2 / cat
    float* bufB    = (float*)carve((size_t)cB * cN * cK * 64 * 4);     // 64 MB: h1
    (void)ws_size; (void)in_sizes; (void)n_in; (void)out_size;

    // ---- stage 1 ----
    IRRTR_knn_self<<<dim3(cN / 256, cB), 256, 0, stream>>>(xy, idx1);
    IRRTR_gradient<<<(cB * cN + 255) / 256, 256, 0, stream>>>(xy, val, idx1, grad);

    // ---- stage 2: feat -> MLP (WMMA) -> neighbor max fused into last GEMM ----
    IRRTR_build_feat<<<(cB * cN + 255) / 256, 256, 0, stream>>>(xy, val, idx1, grad, bufA);
    {
        const int M = cB * cN * cK;                           // 262144 rows
        IRRTR_wmma_gemm<16><<<gemm_blocks(M, 64), 256, 0, stream>>>(
            bufA, sf_w1, sf_b1, bufB, M, 64, 1, 1);
        IRRTR_wmma_gemm<64><<<gemm_blocks(M, 64), 256, 0, stream>>>(
            bufB, sf_w2, sf_b2, bufA, M, 64, 1, 1);
        // pool=16: one M-tile == one point's 16 neighbors -> writes feature directly
        IRRTR_wmma_gemm<64><<<gemm_blocks(M, 128), 256, 0, stream>>>(
            bufA, sf_w3, sf_b3, feature, M, 128, 16, 1);
    }

    // ---- stage 3: grid KNN -> cat -> WMMA GEMM with fused pool-8 -> head ----
    IRRTR_knn_grid<<<dim3(cS2 / 256, cB), 256, 0, stream>>>(xy, idx2);
    IRRTR_build_cat<<<(cB * cS2 * cKP + 255) / 256, 256, 0, stream>>>(xy, val, grad, feature, idx2, bufA);
    {
        const int M = cB * cS2 * cKP;                         // 131072 rows
        // pool=8: one M-tile == two grid nodes -> writes pooled h2 directly
        IRRTR_wmma_gemm<144><<<gemm_blocks(M, 128), 256, 0, stream>>>(
            bufA, rt_w1, rt_b1, h2pool, M, 128, 8, 1);
        const int M2 = cB * cS2;                              // 16384 rows
        IRRTR_wmma_gemm<128><<<gemm_blocks(M2, 128), 256, 0, stream>>>(
            h2pool, cw1, cb1, h3b, M2, 128, 1, 1);
    }
    IRRTR_final<<<(cB * cS2 * cOUT + 255) / 256, 256, 0, stream>>>(h3b, cw2, cb2, out);
}